// RMRM_63763084476814
// MI455X (gfx1250) — compile-verified
//
#include <hip/hip_runtime.h>
#include <math.h>

typedef float v2f __attribute__((ext_vector_type(2)));
typedef float v8f __attribute__((ext_vector_type(8)));

#define NEG_SLOPE 0.2f

// ---------------------------------------------------------------------------
// float atomic max via signed/unsigned int ordering trick (m init = -inf)
// ---------------------------------------------------------------------------
__device__ __forceinline__ void atomicMaxF(float* addr, float v) {
  if (v >= 0.0f) {
    atomicMax((int*)addr, __float_as_int(v));
  } else {
    atomicMin((unsigned int*)addr, __float_as_uint(v));
  }
}

__global__ void fill_kernel(float* __restrict__ p, float v, int n) {
  int i = blockIdx.x * blockDim.x + threadIdx.x;
  if (i < n) p[i] = v;
}

// ---------------------------------------------------------------------------
// C[M,N] = A[M,K] @ B[K,N] with V_WMMA_F32_16X16X4_F32 (wave32).
// One wave per 16x16 tile. blockDim.x=32, blockDim.y = N/16 tiles.
// A 16x4 layout: lane L (L<16): row=L, K=k0,k0+1 ; L>=16: row=L-16, K=k0+2,k0+3
// B 4x16 layout: mirrored (col = lane%16).
// D: VGPR r, lane L -> row = r + (L/16)*8, col = L%16.
// ---------------------------------------------------------------------------
__global__ void gemm_wmma_f32_kernel(const float* __restrict__ A,
                                     const float* __restrict__ B,
                                     float* __restrict__ C,
                                     int M, int N, int K) {
  const int lane = threadIdx.x & 31;
  const int half = lane >> 4;    // 0 or 1
  const int lid  = lane & 15;
  const int row0 = blockIdx.x * 16;
  const int col0 = threadIdx.y * 16;
  const int n = col0 + lid;
  int m = row0 + lid;
  if (m > M - 1) m = M - 1;      // clamp loads (full-EXEC kept for WMMA)

  v8f acc = {};
  const float* arow = A + (size_t)m * K;
  for (int k0 = 0; k0 < K; k0 += 4) {
    const int ka = k0 + half * 2;
    v2f a = *(const v2f*)(arow + ka);
    v2f b;
    b.x = B[(size_t)ka * N + n];
    b.y = B[(size_t)(ka + 1) * N + n];
    acc = __builtin_amdgcn_wmma_f32_16x16x4_f32(
        /*neg_a=*/false, a, /*neg_b=*/false, b,
        /*c_mod=*/(short)0, acc, /*reuse_a=*/false, /*reuse_b=*/false);
  }
#pragma unroll
  for (int r = 0; r < 8; ++r) {
    const int mr = row0 + r + half * 8;
    if (mr < M) C[(size_t)mr * N + n] = acc[r];
  }
}

// ---------------------------------------------------------------------------
// Layer-1 attention (H=8, C=8)
// ---------------------------------------------------------------------------
__global__ void node_att1_kernel(const float* __restrict__ h1,
                                 const float* __restrict__ attS,
                                 const float* __restrict__ attD,
                                 float* __restrict__ as1, float* __restrict__ ad1,
                                 int Nn) {
  int t = blockIdx.x * blockDim.x + threadIdx.x;
  if (t >= Nn * 8) return;
  const int nd = t >> 3, hh = t & 7;
  const float* hp = h1 + (size_t)nd * 64 + hh * 8;
  const float* sp = attS + hh * 8;
  const float* dp = attD + hh * 8;
  float s = 0.f, d = 0.f;
#pragma unroll
  for (int c = 0; c < 8; ++c) { s += hp[c] * sp[c]; d += hp[c] * dp[c]; }
  as1[t] = s; ad1[t] = d;
}

__device__ __forceinline__ void edge_sd(const long long* ei, int E, int e,
                                        int& s, int& d) {
  if (e < E) { s = (int)ei[e]; d = (int)ei[(size_t)E + e]; }
  else       { s = e - E; d = s; }   // appended self-loops
}

__global__ void edge_max1_kernel(const long long* __restrict__ ei, int E, int Nn,
                                 const float* __restrict__ as1,
                                 const float* __restrict__ ad1,
                                 float* __restrict__ m1) {
  int e = blockIdx.x * blockDim.x + threadIdx.x;
  if (e >= E + Nn) return;
  int s, d; edge_sd(ei, E, e, s, d);
#pragma unroll
  for (int hh = 0; hh < 8; ++hh) {
    float x = as1[s * 8 + hh] + ad1[d * 8 + hh];
    x = x > 0.f ? x : NEG_SLOPE * x;
    atomicMaxF(&m1[d * 8 + hh], x);
  }
}

__global__ void edge_sum1_kernel(const long long* __restrict__ ei, int E, int Nn,
                                 const float* __restrict__ as1,
                                 const float* __restrict__ ad1,
                                 const float* __restrict__ m1,
                                 float* __restrict__ s1) {
  int e = blockIdx.x * blockDim.x + threadIdx.x;
  if (e >= E + Nn) return;
  int s, d; edge_sd(ei, E, e, s, d);
#pragma unroll
  for (int hh = 0; hh < 8; ++hh) {
    float x = as1[s * 8 + hh] + ad1[d * 8 + hh];
    x = x > 0.f ? x : NEG_SLOPE * x;
    float mm = m1[d * 8 + hh];
    if (!__builtin_isfinite(mm)) mm = 0.f;
    atomicAdd(&s1[d * 8 + hh], expf(x - mm));
  }
}

__global__ void edge_agg1_kernel(const long long* __restrict__ ei, int E, int Nn,
                                 const float* __restrict__ as1,
                                 const float* __restrict__ ad1,
                                 const float* __restrict__ m1,
                                 const float* __restrict__ s1,
                                 const float* __restrict__ h1,
                                 float* __restrict__ agg1) {
  int t = blockIdx.x * blockDim.x + threadIdx.x;       // (edge, head)
  if (t >= (E + Nn) * 8) return;
  const int e = t >> 3, hh = t & 7;
  int s, d; edge_sd(ei, E, e, s, d);
  float x = as1[s * 8 + hh] + ad1[d * 8 + hh];
  x = x > 0.f ? x : NEG_SLOPE * x;
  float mm = m1[d * 8 + hh];
  if (!__builtin_isfinite(mm)) mm = 0.f;
  const float ex = expf(x - mm);
  const float ss = s1[d * 8 + hh];
  const float alpha = ex / (ss > 0.f ? ss : 1.f);
  const float* hp = h1 + (size_t)s * 64 + hh * 8;
  float* op = agg1 + (size_t)d * 64 + hh * 8;
#pragma unroll
  for (int c = 0; c < 8; ++c) atomicAdd(&op[c], alpha * hp[c]);
}

__global__ void bias_elu_kernel(float* __restrict__ a, const float* __restrict__ b,
                                int Nn) {
  int t = blockIdx.x * blockDim.x + threadIdx.x;
  if (t >= Nn * 64) return;
  float v = a[t] + b[t & 63];
  a[t] = v > 0.f ? v : (expf(v) - 1.f);
}

// ---------------------------------------------------------------------------
// Layer-2 attention (H=1, C=128)
// ---------------------------------------------------------------------------
__global__ void node_att2_kernel(const float* __restrict__ h2,
                                 const float* __restrict__ aS,
                                 const float* __restrict__ aD,
                                 float* __restrict__ as2, float* __restrict__ ad2,
                                 int Nn) {
  int nd = blockIdx.x * blockDim.x + threadIdx.x;
  if (nd >= Nn) return;
  const float* hp = h2 + (size_t)nd * 128;
  float s = 0.f, d = 0.f;
#pragma unroll 8
  for (int j = 0; j < 128; ++j) { float v = hp[j]; s += v * aS[j]; d += v * aD[j]; }
  as2[nd] = s; ad2[nd] = d;
}

__global__ void edge_max2_kernel(const long long* __restrict__ ei, int E, int Nn,
                                 const float* __restrict__ as2,
                                 const float* __restrict__ ad2,
                                 float* __restrict__ m2) {
  int e = blockIdx.x * blockDim.x + threadIdx.x;
  if (e >= E + Nn) return;
  int s, d; edge_sd(ei, E, e, s, d);
  float x = as2[s] + ad2[d];
  x = x > 0.f ? x : NEG_SLOPE * x;
  atomicMaxF(&m2[d], x);
}

__global__ void edge_sum2_kernel(const long long* __restrict__ ei, int E, int Nn,
                                 const float* __restrict__ as2,
                                 const float* __restrict__ ad2,
                                 const float* __restrict__ m2,
                                 float* __restrict__ s2) {
  int e = blockIdx.x * blockDim.x + threadIdx.x;
  if (e >= E + Nn) return;
  int s, d; edge_sd(ei, E, e, s, d);
  float x = as2[s] + ad2[d];
  x = x > 0.f ? x : NEG_SLOPE * x;
  float mm = m2[d];
  if (!__builtin_isfinite(mm)) mm = 0.f;
  atomicAdd(&s2[d], expf(x - mm));
}

__global__ void edge_agg2_kernel(const long long* __restrict__ ei, int E, int Nn,
                                 const float* __restrict__ as2,
                                 const float* __restrict__ ad2,
                                 const float* __restrict__ m2,
                                 const float* __restrict__ s2,
                                 const float* __restrict__ h2,
                                 float* __restrict__ out) {
  int t = blockIdx.x * blockDim.x + threadIdx.x;       // (edge, quarter)
  if (t >= (E + Nn) * 4) return;
  const int e = t >> 2, q = t & 3;
  int s, d; edge_sd(ei, E, e, s, d);
  float x = as2[s] + ad2[d];
  x = x > 0.f ? x : NEG_SLOPE * x;
  float mm = m2[d];
  if (!__builtin_isfinite(mm)) mm = 0.f;
  const float ex = expf(x - mm);
  const float ss = s2[d];
  const float alpha = ex / (ss > 0.f ? ss : 1.f);
  const float* hp = h2 + (size_t)s * 128 + q * 32;
  float* op = out + (size_t)d * 128 + q * 32;
#pragma unroll
  for (int c = 0; c < 32; ++c) atomicAdd(&op[c], alpha * hp[c]);
}

__global__ void bias2_kernel(float* __restrict__ out, const float* __restrict__ b,
                             int Nn) {
  int t = blockIdx.x * blockDim.x + threadIdx.x;
  if (t >= Nn * 128) return;
  out[t] += b[t & 127];
}

// ---------------------------------------------------------------------------
extern "C" void kernel_launch(void* const* d_in, const int* in_sizes, int n_in,
                              void* d_out, int out_size, void* d_ws, size_t ws_size,
                              hipStream_t stream) {
  const float*     x   = (const float*)d_in[0];
  const long long* ei  = (const long long*)d_in[1];   // int64 [2,E]
  const float*     W1  = (const float*)d_in[2];       // [128,64]
  const float*     aS1 = (const float*)d_in[3];       // [8,8]
  const float*     aD1 = (const float*)d_in[4];
  const float*     b1  = (const float*)d_in[5];       // [64]
  const float*     W2  = (const float*)d_in[6];       // [64,128]
  const float*     aS2 = (const float*)d_in[7];       // [128]
  const float*     aD2 = (const float*)d_in[8];
  const float*     b2  = (const float*)d_in[9];       // [128]
  float*           out = (float*)d_out;

  const int F  = 128, HC1 = 64;
  const int Nn = in_sizes[0] / F;       // 50000 (multiple of 16)
  const int E  = in_sizes[1] / 2;       // 800000
  const int EE = E + Nn;                // with self-loops

  // workspace layout (floats)
  float* ws = (float*)d_ws;
  size_t off = 0;
  float* h1   = ws + off; off += (size_t)Nn * HC1;    // [N,64]
  float* as1  = ws + off; off += (size_t)Nn * 8;
  float* ad1  = ws + off; off += (size_t)Nn * 8;
  float* m1   = ws + off; off += (size_t)Nn * 8;
  float* s1   = ws + off; off += (size_t)Nn * 8;
  float* agg1 = ws + off; off += (size_t)Nn * HC1;    // becomes elu(h1+b1)
  float* h2   = ws + off; off += (size_t)Nn * F;      // [N,128]
  float* as2  = ws + off; off += (size_t)Nn;
  float* ad2  = ws + off; off += (size_t)Nn;
  float* m2   = ws + off; off += (size_t)Nn;
  float* s2   = ws + off; off += (size_t)Nn;

  const int TB = 256;
  auto cdiv = [](int a, int b) { return (a + b - 1) / b; };
  const int mtiles = cdiv(Nn, 16);

  // ---- init accumulators (d_out/ws are poisoned by harness) ----
  fill_kernel<<<cdiv(Nn * 8, TB), TB, 0, stream>>>(m1, -INFINITY, Nn * 8);
  fill_kernel<<<cdiv(Nn * 8, TB), TB, 0, stream>>>(s1, 0.f, Nn * 8);
  fill_kernel<<<cdiv(Nn * HC1, TB), TB, 0, stream>>>(agg1, 0.f, Nn * HC1);
  fill_kernel<<<cdiv(Nn, TB), TB, 0, stream>>>(m2, -INFINITY, Nn);
  fill_kernel<<<cdiv(Nn, TB), TB, 0, stream>>>(s2, 0.f, Nn);
  fill_kernel<<<cdiv(Nn * F, TB), TB, 0, stream>>>(out, 0.f, Nn * F);

  // ---- layer 1 ----
  gemm_wmma_f32_kernel<<<dim3(mtiles), dim3(32, HC1 / 16), 0, stream>>>(
      x, W1, h1, Nn, HC1, F);                                   // h1 = x @ W1
  node_att1_kernel<<<cdiv(Nn * 8, TB), TB, 0, stream>>>(h1, aS1, aD1, as1, ad1, Nn);
  edge_max1_kernel<<<cdiv(EE, TB), TB, 0, stream>>>(ei, E, Nn, as1, ad1, m1);
  edge_sum1_kernel<<<cdiv(EE, TB), TB, 0, stream>>>(ei, E, Nn, as1, ad1, m1, s1);
  edge_agg1_kernel<<<cdiv(EE * 8, TB), TB, 0, stream>>>(ei, E, Nn, as1, ad1, m1, s1,
                                                        h1, agg1);
  bias_elu_kernel<<<cdiv(Nn * HC1, TB), TB, 0, stream>>>(agg1, b1, Nn);

  // ---- layer 2 ----
  gemm_wmma_f32_kernel<<<dim3(mtiles), dim3(32, F / 16), 0, stream>>>(
      agg1, W2, h2, Nn, F, HC1);                                // h2 = h1act @ W2
  node_att2_kernel<<<cdiv(Nn, TB), TB, 0, stream>>>(h2, aS2, aD2, as2, ad2, Nn);
  edge_max2_kernel<<<cdiv(EE, TB), TB, 0, stream>>>(ei, E, Nn, as2, ad2, m2);
  edge_sum2_kernel<<<cdiv(EE, TB), TB, 0, stream>>>(ei, E, Nn, as2, ad2, m2, s2);
  edge_agg2_kernel<<<cdiv(EE * 4, TB), TB, 0, stream>>>(ei, E, Nn, as2, ad2, m2, s2,
                                                        h2, out);
  bias2_kernel<<<cdiv(Nn * F, TB), TB, 0, stream>>>(out, b2, Nn);
}